// MultiHeadProductKeyRouter_9620726743479
// MI455X (gfx1250) — compile-verified
//
#include <hip/hip_runtime.h>
#include <cstddef>

typedef float v2f __attribute__((ext_vector_type(2)));
typedef float v4f __attribute__((ext_vector_type(4)));
typedef float v8f __attribute__((ext_vector_type(8)));

#define DIM   2048
#define SK    64
#define NH    4
#define TOPK  8
#define NEG_INF (-3.402823466e38f)

// ---------------------------------------------------------------------------
// Kernel 0: fold heads into reduced weight matrices with the 1/H mean factor.
// W1r[k1][d] = 0.25 * sum_h W1[(h*64+k1)*D + d]   (64 x 2048 each)
// ---------------------------------------------------------------------------
__global__ void reduce_weights(const float* __restrict__ W1,
                               const float* __restrict__ W2,
                               float* __restrict__ W1r,
                               float* __restrict__ W2r) {
    int i = blockIdx.x * blockDim.x + threadIdx.x;   // 0 .. 64*2048-1
    if (i >= SK * DIM) return;
    int k1 = i / DIM, d = i - k1 * DIM;
    float s1 = 0.f, s2 = 0.f;
#pragma unroll
    for (int h = 0; h < NH; ++h) {
        s1 += W1[(size_t)(h * SK + k1) * DIM + d];
        s2 += W2[(size_t)(h * SK + k1) * DIM + d];
    }
    W1r[i] = 0.25f * s1;
    W2r[i] = 0.25f * s2;
}

// ---------------------------------------------------------------------------
// Kernel 1: A = x @ W1r^T, B = x @ W2r^T via V_WMMA_F32_16X16X4_F32.
// One wave handles 16 tokens x 64 cols for BOTH matrices (8 tiles of 16x16).
// f32 WMMA fragment layout (ISA 7.12.2):
//   A 16x4 : lane L -> M=L%16, VGPR{0,1} = K = (L/16)*2 + {0,1}
//   B 4x16 : lane L -> N=L%16, VGPR{0,1} = K = (L/16)*2 + {0,1}
//   C/D    : lane L -> N=L%16, VGPR r -> M = r + 8*(L/16)
// A float4 load per lane supplies two WMMA K-steps (K permutation is safe
// because A and B fragments use the identical lane->K mapping).
// x is streamed with NT loads (read exactly once); W1r/W2r stay RT so the
// 1 MB of reduced weights remains L2-resident for all 1024 waves.
// ---------------------------------------------------------------------------
__global__ void __launch_bounds__(128)
pk_gemm(const float* __restrict__ X,
        const float* __restrict__ W1r,
        const float* __restrict__ W2r,
        float* __restrict__ Asc,
        float* __restrict__ Bsc) {
    const int lane = threadIdx.x & 31;
    const int wave = threadIdx.x >> 5;
    const int tb   = blockIdx.x * 4 + wave;     // 16-token block index
    const int m    = lane & 15;
    const int g    = lane >> 4;                 // K-pair group (0 or 1)

    const float* xrow = X + (size_t)(tb * 16 + m) * DIM + g * 4;
    const float* wrow[8];
#pragma unroll
    for (int j = 0; j < 8; ++j) {
        const float* base = (j < 4) ? W1r : W2r;
        int n = (j & 3) * 16 + m;
        wrow[j] = base + (size_t)n * DIM + g * 4;
    }

    v8f zero = {};
    v8f acc[8];
#pragma unroll
    for (int j = 0; j < 8; ++j) acc[j] = zero;

    for (int k = 0; k < DIM; k += 8) {
        v4f xa = __builtin_nontemporal_load((const v4f*)(xrow + k));
        v4f wb[8];
#pragma unroll
        for (int j = 0; j < 8; ++j) wb[j] = *(const v4f*)(wrow[j] + k);
#pragma unroll
        for (int j = 0; j < 8; ++j) {
            v2f a0 = {xa.x, xa.y};
            v2f b0 = {wb[j].x, wb[j].y};
            acc[j] = __builtin_amdgcn_wmma_f32_16x16x4_f32(
                false, a0, false, b0, (short)0, acc[j], false, false);
            v2f a1 = {xa.z, xa.w};
            v2f b1 = {wb[j].z, wb[j].w};
            acc[j] = __builtin_amdgcn_wmma_f32_16x16x4_f32(
                false, a1, false, b1, (short)0, acc[j], false, false);
        }
    }

    // Store: lane L holds N=L%16, rows M = r + 8*(L/16)
    const int n    = lane & 15;
    const int half = lane >> 4;
#pragma unroll
    for (int j = 0; j < 8; ++j) {
        float* outp = (j < 4) ? Asc : Bsc;
        int ct = j & 3;
#pragma unroll
        for (int r = 0; r < 8; ++r) {
            int mm = r + 8 * half;
            outp[(size_t)(tb * 16 + mm) * SK + ct * 16 + n] = acc[j][r];
        }
    }
}

// ---------------------------------------------------------------------------
// Kernel 2: per token — scores = a[k1]+b[k2] (write all 4096 with NT stores:
// 256 MB streamed once, never re-read — keep it out of L2), top-8 with
// lowest-index tie-break (matches jax.lax.top_k), softmax gates.
// One 128-thread block per token.
// ---------------------------------------------------------------------------
__global__ void __launch_bounds__(128)
pk_scores_topk(const float* __restrict__ Asc,
               const float* __restrict__ Bsc,
               float* __restrict__ out_idx,
               float* __restrict__ out_gates,
               float* __restrict__ out_scores) {
    const int t   = blockIdx.x;
    const int tid = threadIdx.x;

    __shared__ float a[SK];
    __shared__ float b[SK];
    __shared__ float rv[128];
    __shared__ int   rm[128];
    __shared__ float topv[TOPK];
    __shared__ int   topi[TOPK];

    if (tid < SK)            a[tid]       = Asc[(size_t)t * SK + tid];
    else if (tid < 2 * SK)   b[tid - SK]  = Bsc[(size_t)t * SK + (tid - SK)];
    __syncthreads();

    // per-thread top-8 (sorted descending; strict '>' keeps earlier index on tie)
    float lv[TOPK];
    int   li[TOPK];
#pragma unroll
    for (int r = 0; r < TOPK; ++r) { lv[r] = NEG_INF; li[r] = 0; }

    float* sc = out_scores + (size_t)t * (SK * SK);
#pragma unroll 4
    for (int i = 0; i < 32; ++i) {
        int k = tid + 128 * i;                      // coalesced across threads
        float v = a[k >> 6] + b[k & 63];
        __builtin_nontemporal_store(v, &sc[k]);
        if (v > lv[TOPK - 1]) {
            int p = TOPK - 1;
            while (p > 0 && lv[p - 1] < v) {
                lv[p] = lv[p - 1]; li[p] = li[p - 1]; --p;
            }
            lv[p] = v; li[p] = k;
        }
    }

    // 8 rounds of block argmax over each thread's current best candidate
    int ptr = 0;
    for (int r = 0; r < TOPK; ++r) {
        float cv = (ptr < TOPK) ? lv[ptr] : NEG_INF;
        int   ck = (ptr < TOPK) ? li[ptr] : 0x7FFF;
        rv[tid] = cv;
        rm[tid] = (ck << 7) | tid;                  // smaller meta = smaller k
        __syncthreads();
        for (int s = 64; s > 0; s >>= 1) {
            if (tid < s) {
                float v2 = rv[tid + s]; int m2 = rm[tid + s];
                if (v2 > rv[tid] || (v2 == rv[tid] && m2 < rm[tid])) {
                    rv[tid] = v2; rm[tid] = m2;
                }
            }
            __syncthreads();
        }
        float wv = rv[0]; int wm = rm[0];
        if (tid == 0) { topv[r] = wv; topi[r] = wm >> 7; }
        if ((wm & 127) == tid) ++ptr;               // winner advances its list
        __syncthreads();
    }

    if (tid == 0) {
        float mx = topv[0];
        float e[TOPK];
        float sum = 0.f;
#pragma unroll
        for (int r = 0; r < TOPK; ++r) { e[r] = __expf(topv[r] - mx); sum += e[r]; }
        float inv = 1.f / sum;
#pragma unroll
        for (int r = 0; r < TOPK; ++r) {
            out_idx[(size_t)t * TOPK + r]   = (float)topi[r];
            out_gates[(size_t)t * TOPK + r] = e[r] * inv;
        }
    }
}

// ---------------------------------------------------------------------------
extern "C" void kernel_launch(void* const* d_in, const int* in_sizes, int n_in,
                              void* d_out, int out_size, void* d_ws, size_t ws_size,
                              hipStream_t stream) {
    const float* x  = (const float*)d_in[0];   // (B,S,2048) fp32
    const float* W1 = (const float*)d_in[1];   // (256,2048) fp32
    const float* W2 = (const float*)d_in[2];   // (256,2048) fp32

    const int T = in_sizes[0] / DIM;           // total tokens (16384)

    float* ws   = (float*)d_ws;
    float* W1r  = ws;                          //  64*2048
    float* W2r  = W1r + SK * DIM;              //  64*2048
    float* Asc  = W2r + SK * DIM;              //  T*64
    float* Bsc  = Asc + (size_t)T * SK;        //  T*64   (total ~9 MB)

    float* out        = (float*)d_out;
    float* out_idx    = out;                           // (T,8) idx as float
    float* out_gates  = out + (size_t)T * TOPK;        // (T,8)
    float* out_scores = out + (size_t)2 * T * TOPK;    // (T,4096)

    reduce_weights<<<(SK * DIM + 255) / 256, 256, 0, stream>>>(W1, W2, W1r, W2r);
    pk_gemm<<<T / 64, 128, 0, stream>>>(x, W1r, W2r, Asc, Bsc);
    pk_scores_topk<<<T, 128, 0, stream>>>(Asc, Bsc, out_idx, out_gates, out_scores);
}